// LightGCNRecommender_41231686041681
// MI455X (gfx1250) — compile-verified
//
#include <hip/hip_runtime.h>
#include <hip/hip_bf16.h>
#include <stdint.h>

#define WPB   8     // waves per 256-thread block (wave32)
#define CHUNK 32    // edges staged per async chunk (one b64 per lane)

struct __align__(8) EdgeEnt { unsigned src; float w; };

typedef __attribute__((ext_vector_type(2))) float v2f;
typedef __attribute__((ext_vector_type(8))) float v8f;

__device__ __forceinline__ unsigned umin32(unsigned a, unsigned b) { return a < b ? a : b; }

// ---- CDNA5 async global->LDS staging (inline asm: portable across toolchains) ----
__device__ __forceinline__ void async_ld_b64(unsigned lds_off, const void* g) {
  asm volatile("global_load_async_to_lds_b64 %0, %1, off" :: "v"(lds_off), "v"(g) : "memory");
}
__device__ __forceinline__ void wait_async0() { asm volatile("s_wait_asynccnt 0x0" ::: "memory"); }
__device__ __forceinline__ void wait_async1() { asm volatile("s_wait_asynccnt 0x1" ::: "memory"); }

// ---------------- setup kernels ----------------
__global__ __launch_bounds__(256) void k_zero_u32(unsigned* __restrict__ p, int n) {
  int i = blockIdx.x * 256 + threadIdx.x;
  if (i < n) p[i] = 0u;
}

__global__ __launch_bounds__(256) void k_count(const int* __restrict__ dst, unsigned* __restrict__ deg, int E) {
  int e = blockIdx.x * 256 + threadIdx.x;
  if (e < E) atomicAdd(&deg[dst[e]], 1u);
}

__global__ __launch_bounds__(256) void k_dinv(const unsigned* __restrict__ deg, float* __restrict__ dinv, int n) {
  int i = blockIdx.x * 256 + threadIdx.x;
  if (i < n) {
    unsigned d = deg[i];
    dinv[i] = d ? rsqrtf((float)d) : 0.0f;  // matches where(deg>0, rsqrt(max(deg,1)), 0)
  }
}

// single-block chunked exclusive scan over degree counts -> row offsets + scatter cursors
__global__ __launch_bounds__(256) void k_scan(const unsigned* __restrict__ cnt,
                                              unsigned* __restrict__ rowoff,
                                              unsigned* __restrict__ cursor, int n) {
  __shared__ unsigned buf[256];
  __shared__ unsigned carry;
  int t = threadIdx.x;
  if (t == 0) carry = 0;
  __syncthreads();
  for (int base = 0; base < n; base += 256) {
    unsigned v = (base + t < n) ? cnt[base + t] : 0u;
    buf[t] = v;
    __syncthreads();
    for (int off = 1; off < 256; off <<= 1) {
      unsigned add = (t >= off) ? buf[t - off] : 0u;
      __syncthreads();
      buf[t] += add;
      __syncthreads();
    }
    unsigned excl = carry + buf[t] - v;
    if (base + t < n) { rowoff[base + t] = excl; cursor[base + t] = excl; }
    __syncthreads();
    if (t == 255) carry += buf[255];
    __syncthreads();
  }
  if (t == 0) rowoff[n] = carry;
}

__global__ __launch_bounds__(256) void k_scatter(const int* __restrict__ src, const int* __restrict__ dst,
                                                 const float* __restrict__ dinv, unsigned* __restrict__ cursor,
                                                 EdgeEnt* __restrict__ csr, int E) {
  int e = blockIdx.x * 256 + threadIdx.x;
  if (e < E) {
    int s = src[e], d = dst[e];
    unsigned pos = atomicAdd(&cursor[d], 1u);
    EdgeEnt en; en.src = (unsigned)s; en.w = dinv[s] * dinv[d];
    csr[pos] = en;
  }
}

__global__ __launch_bounds__(256) void k_copy4(const float4* __restrict__ a, float4* __restrict__ b, int n4) {
  int i = blockIdx.x * 256 + threadIdx.x;
  if (i < n4) b[i] = a[i];
}

// ---------------- atomic-free CSR SpMM: one wave per dst row ----------------
// x_next[row] = sum_e w_e * x[src_e];  out[row] += x_next[row]
__global__ __launch_bounds__(256) void k_spmm(const float* __restrict__ x, const EdgeEnt* __restrict__ csr,
                                              const unsigned* __restrict__ rowoff,
                                              float* __restrict__ xn, float* __restrict__ out, int n) {
  __shared__ EdgeEnt stage[WPB][2][CHUNK];   // per-wave double-buffered edge staging (4 KB)
  int lane = threadIdx.x & 31;
  int wv   = threadIdx.x >> 5;
  int row  = blockIdx.x * WPB + wv;
  if (row >= n) return;                      // wave-uniform

  unsigned beg = rowoff[row], end = rowoff[row + 1];
  unsigned degr = end - beg;
  float2 acc; acc.x = 0.0f; acc.y = 0.0f;

  unsigned nch = (degr + (CHUNK - 1)) >> 5;
  if (nch) {
    if ((unsigned)lane < degr) {
      unsigned lds = (unsigned)(uintptr_t)&stage[wv][0][lane];
      async_ld_b64(lds, (const void*)(csr + beg + lane));
    }
  }
  int bb = 0;
  for (unsigned c = 0; c < nch; ++c) {
    unsigned cb = c << 5;
    if (c + 1 < nch) {
      unsigned nb = cb + CHUNK;
      unsigned rem = degr - nb;
      if ((unsigned)lane < rem) {
        unsigned lds = (unsigned)(uintptr_t)&stage[wv][bb ^ 1][lane];
        async_ld_b64(lds, (const void*)(csr + beg + nb + lane));
      }
      wait_async1();   // chunk c done; chunk c+1 stays in flight
    } else {
      wait_async0();
    }
    unsigned cnt = umin32(CHUNK, degr - cb);
    const EdgeEnt* st = &stage[wv][bb][0];
    unsigned j = 0;
    for (; j + 4 <= cnt; j += 4) {           // 4 independent gathers in flight
      EdgeEnt e0 = st[j], e1 = st[j + 1], e2 = st[j + 2], e3 = st[j + 3];
      float2 x0 = *(const float2*)(x + (size_t)e0.src * 64 + (lane << 1));
      float2 x1 = *(const float2*)(x + (size_t)e1.src * 64 + (lane << 1));
      float2 x2 = *(const float2*)(x + (size_t)e2.src * 64 + (lane << 1));
      float2 x3 = *(const float2*)(x + (size_t)e3.src * 64 + (lane << 1));
      acc.x = fmaf(e0.w, x0.x, fmaf(e1.w, x1.x, fmaf(e2.w, x2.x, fmaf(e3.w, x3.x, acc.x))));
      acc.y = fmaf(e0.w, x0.y, fmaf(e1.w, x1.y, fmaf(e2.w, x2.y, fmaf(e3.w, x3.y, acc.y))));
    }
    for (; j < cnt; ++j) {
      EdgeEnt e = st[j];
      float2 xv = *(const float2*)(x + (size_t)e.src * 64 + (lane << 1));
      acc.x = fmaf(e.w, xv.x, acc.x);
      acc.y = fmaf(e.w, xv.y, acc.y);
    }
    bb ^= 1;
  }

  size_t o = (size_t)row * 64 + (lane << 1);
  *(float2*)(xn + o) = acc;                  // full row written (zeros for isolated nodes)
  float2 ov = *(const float2*)(out + o);
  ov.x += acc.x; ov.y += acc.y;
  *(float2*)(out + o) = ov;
}

// ---------------- finalize: out *= 0.25 via D = (0.25*I) x out_tile (WMMA f32 16x16x4) ----------------
__global__ __launch_bounds__(256) void k_finalize(float* __restrict__ out, int nwaves) {
  int lane = threadIdx.x & 31;
  int wv   = threadIdx.x >> 5;
  int wid  = blockIdx.x * WPB + wv;
  if (wid >= nwaves) return;                 // wave-uniform; active waves keep EXEC all-ones
  int tr = wid >> 2;                         // 16-node tile
  int tc = wid & 3;                          // 16-dim tile (64/16)
  int half = lane >> 4;
  int nn   = lane & 15;
  size_t base = (size_t)tr * (16 * 64) + (size_t)tc * 16;
#if __has_builtin(__builtin_amdgcn_wmma_f32_16x16x4_f32)
  v8f c = {};
#pragma unroll
  for (int k = 0; k < 4; ++k) {
    int r0 = 4 * k + (half ? 2 : 0);         // K slots: V0 = K(2*half), V1 = K(2*half)+1
    v2f a, b;
    a.x = (nn == r0)     ? 0.25f : 0.0f;     // A = 0.25*I[:, 4k:4k+4], lane = M row
    a.y = (nn == r0 + 1) ? 0.25f : 0.0f;
    b.x = out[base + (size_t)(r0)     * 64 + nn];   // B = out_tile[4k:4k+4, :]
    b.y = out[base + (size_t)(r0 + 1) * 64 + nn];
    c = __builtin_amdgcn_wmma_f32_16x16x4_f32(false, a, false, b, (short)0, c, false, false);
  }
#pragma unroll
  for (int v = 0; v < 8; ++v)               // D: VGPR v -> row v + 8*half, col nn
    out[base + (size_t)(v + (half ? 8 : 0)) * 64 + nn] = c[v];
#else
  for (int r = 0; r < 8; ++r) {
    size_t i = base + (size_t)(r + (half ? 8 : 0)) * 64 + nn;
    out[i] *= 0.25f;
  }
#endif
}

__global__ __launch_bounds__(256) void k_scale_tail(float* __restrict__ out, size_t start, size_t count) {
  size_t i = (size_t)blockIdx.x * 256 + threadIdx.x;
  if (i < count) out[start + i] *= 0.25f;
}

extern "C" void kernel_launch(void* const* d_in, const int* in_sizes, int n_in,
                              void* d_out, int out_size, void* d_ws, size_t ws_size,
                              hipStream_t stream) {
  (void)n_in; (void)out_size; (void)ws_size;
  const float* emb = (const float*)d_in[0];
  const int*   ei  = (const int*)d_in[1];   // [2, E] flat: src rows then dst rows
  const int D = 64;
  int n = in_sizes[0] / D;                  // 100000
  int E = in_sizes[1] / 2;                  // 3200000
  float* out = (float*)d_out;

  // workspace carve-up (~78.5 MB total)
  char* ws = (char*)d_ws;
  size_t off = 0;
  auto take = [&](size_t bytes) -> void* {
    void* p = ws + off;
    off = (off + bytes + 15) & ~(size_t)15;
    return p;
  };
  unsigned* deg    = (unsigned*)take((size_t)n * 4);
  float*    dinv   = (float*)   take((size_t)n * 4);
  unsigned* rowoff = (unsigned*)take((size_t)(n + 1) * 4);
  unsigned* cursor = (unsigned*)take((size_t)n * 4);
  EdgeEnt*  csr    = (EdgeEnt*) take((size_t)E * 8);
  float*    xA     = (float*)   take((size_t)n * D * 4);
  float*    xB     = (float*)   take((size_t)n * D * 4);

  int gb_n = (n + 255) / 256;
  int gb_E = (E + 255) / 256;

  // degree + normalization
  k_zero_u32<<<gb_n, 256, 0, stream>>>(deg, n);
  k_count  <<<gb_E, 256, 0, stream>>>(ei + E, deg, E);
  k_dinv   <<<gb_n, 256, 0, stream>>>(deg, dinv, n);
  // COO -> CSR with fused edge weights (built once, reused for all 3 layers)
  k_scan   <<<1,    256, 0, stream>>>(deg, rowoff, cursor, n);
  k_scatter<<<gb_E, 256, 0, stream>>>(ei, ei + E, dinv, cursor, csr, E);

  // out = emb
  int n4 = n * D / 4;
  k_copy4<<<(n4 + 255) / 256, 256, 0, stream>>>((const float4*)emb, (float4*)out, n4);

  // 3 LGConv layers, atomic-free, accumulating into out
  int gb_rows = (n + WPB - 1) / WPB;
  k_spmm<<<gb_rows, 256, 0, stream>>>(emb, csr, rowoff, xA, out, n);
  k_spmm<<<gb_rows, 256, 0, stream>>>(xA,  csr, rowoff, xB, out, n);
  k_spmm<<<gb_rows, 256, 0, stream>>>(xB,  csr, rowoff, xA, out, n);

  // out *= 0.25 (WMMA tiles + scalar tail for generality)
  int ntile_rows = n / 16;
  int nwaves = ntile_rows * 4;
  if (nwaves)
    k_finalize<<<(nwaves + WPB - 1) / WPB, 256, 0, stream>>>(out, nwaves);
  int tail_rows = n - ntile_rows * 16;
  if (tail_rows) {
    size_t start = (size_t)ntile_rows * 16 * D;
    size_t count = (size_t)tail_rows * D;
    k_scale_tail<<<(int)((count + 255) / 256), 256, 0, stream>>>(out, start, count);
  }
}